// GNN_RefinedNet_52123723105072
// MI455X (gfx1250) — compile-verified
//
#include <hip/hip_runtime.h>

typedef __attribute__((ext_vector_type(2))) float v2f;
typedef __attribute__((ext_vector_type(8))) float v8f;

#define HH    128
#define WW    208
#define CINC  256
#define OUTC  64
#define NKPT  15
#define NPRED 10
#define NVIEW 2

// ---------------------------------------------------------------------------
// Kernel A: gathered 1x1-conv + BN + ReLU + 3x3 max-pool at each keypoint.
// One block (128 thr = 4 wave32) per (v,n,k). GEMM: 16 rows (9 window pixels,
// padded) x 64 out-ch x 256 K via V_WMMA_F32_16X16X4_F32.
// ---------------------------------------------------------------------------
__global__ __launch_bounds__(128) void kp_conv_pool(
    const float* __restrict__ pre_feature,   // (V,1,256,H,W)
    const int*   __restrict__ pred_xy,       // (V,1,N,KPT,2)
    const float* __restrict__ conv_w,        // (64,256)
    const float* __restrict__ conv_b,        // (64,)
    const float* __restrict__ bn_gamma,
    const float* __restrict__ bn_beta,
    const float* __restrict__ bn_mean,
    const float* __restrict__ bn_var,
    float* __restrict__ pooled)              // (V,N,KPT,64)
{
    __shared__ float Xs[16][CINC + 4];       // +4 pad: stride 260 breaks bank conflicts
    const int bid = blockIdx.x;              // flattened (v,n,k): 0..299
    const int v   = bid / (NPRED * NKPT);
    const int t   = threadIdx.x;

    const int xs = pred_xy[bid * 2 + 0];
    const int ys = pred_xy[bid * 2 + 1];

    // Stage A: 16 rows x 256 channels; row m<9 is window pixel (dy,dx)=(m/3-1,m%3-1)
    const float* pf = pre_feature + (size_t)v * CINC * HH * WW;
    for (int e = t; e < 16 * CINC; e += 128) {
        const int m = e >> 8;
        const int c = e & (CINC - 1);
        float val = 0.f;
        if (m < 9) {
            const int y = ys + (m / 3) - 1;
            const int x = xs + (m % 3) - 1;
            if (y >= 0 && y < HH && x >= 0 && x < WW)
                val = pf[(size_t)c * (HH * WW) + y * WW + x];
        }
        Xs[m][c] = val;
    }
    __syncthreads();

    const int wave = t >> 5;
    const int lane = t & 31;
    const int hi   = lane >> 4;              // half-wave selector
    const int l16  = lane & 15;              // M (A-row) and N (B-col) within tile
    const int ncol = wave * 16 + l16;        // global out-channel
    const int koff = hi * 2;                 // lanes 16-31 hold K+2,K+3

    const float* wrow = conv_w + (size_t)ncol * CINC;
    v8f acc = {0.f, 0.f, 0.f, 0.f, 0.f, 0.f, 0.f, 0.f};
    #pragma unroll 8
    for (int k0 = 0; k0 < CINC; k0 += 4) {
        v2f a, b;
        a.x = Xs[l16][k0 + koff];            // A[m][k], A[m][k+1]
        a.y = Xs[l16][k0 + koff + 1];
        b.x = wrow[k0 + koff];               // B[k][n] = conv_w[n][k]
        b.y = wrow[k0 + koff + 1];
        acc = __builtin_amdgcn_wmma_f32_16x16x4_f32(
            false, a, false, b, (short)0, acc, false, false);
    }

    // bias + BN + ReLU, then max over the (clipped) 3x3 window.
    const float cb = conv_b[ncol];
    const float sc = bn_gamma[ncol] / sqrtf(bn_var[ncol] + 1e-5f);
    const float sh = bn_beta[ncol] - bn_mean[ncol] * sc;

    float mx = 0.f;                          // ReLU outputs >= 0, so 0 is neutral
    #pragma unroll
    for (int r = 0; r < 8; ++r) {
        const int row = r + hi * 8;          // D layout: lane l, vgpr r -> M = r + 8*(l>=16)
        const float val = fmaxf((acc[r] + cb) * sc + sh, 0.f);
        bool rv = false;
        if (row < 9) {
            const int y = ys + (row / 3) - 1;
            const int x = xs + (row % 3) - 1;
            rv = (y >= 0 && y < HH && x >= 0 && x < WW);
        }
        if (rv) mx = fmaxf(mx, val);
    }
    mx = fmaxf(mx, __shfl_xor(mx, 16, 32));  // combine the two half-wave row groups
    if (hi == 0)
        pooled[(size_t)bid * OUTC + ncol] = mx;
}

// ---------------------------------------------------------------------------
// Kernel B: per-sample node build (pe encoding + valid-weighted view average)
// followed by the 4-layer per-keypoint GNN with the fixed skeleton Laplacian.
// LG[i][j] = adj(i,j) / sqrt(deg_i*deg_j), baked as bitmasks + 1/sqrt(deg).
// ---------------------------------------------------------------------------
__constant__ unsigned kAdj[NKPT] = {
    0x020Fu, 0x0003u, 0x1045u, 0x0019u, 0x0038u, 0x0030u, 0x00C4u, 0x01C0u,
    0x0180u, 0x0601u, 0x0E00u, 0x0C00u, 0x3004u, 0x7000u, 0x6000u};
__constant__ float kDinv[NKPT] = {
    0.44721360f, 0.70710678f, 0.5f,        0.57735027f, 0.57735027f,
    0.70710678f, 0.57735027f, 0.57735027f, 0.70710678f, 0.57735027f,
    0.57735027f, 0.70710678f, 0.57735027f, 0.57735027f, 0.70710678f};

__global__ __launch_bounds__(128) void kp_gnn(
    const float* __restrict__ pooled,        // (V,N,KPT,64)
    const float* __restrict__ valid,         // (V,1,N,KPT,1)
    const float* __restrict__ pro_inv,       // (V,4,3)
    const float* __restrict__ cam_enc,       // (3,2)
    const float* __restrict__ g0,            // (15,128,64)
    const float* __restrict__ g1,            // (15,64,32)
    const float* __restrict__ g2,            // (15,32,16)
    const float* __restrict__ g3,            // (15,16,3)
    float* __restrict__ out)                 // (1,N,KPT,3)
{
    __shared__ float xb[NKPT][128];
    __shared__ float tb[NKPT][128];
    __shared__ float pe[NVIEW][4][2];
    __shared__ float vs[NVIEW][NKPT];
    const int n = blockIdx.x;                // sample 0..9
    const int t = threadIdx.x;

    if (t < 16) {                            // pe = pro_inv[v] @ cam_enc  -> (4,2) per view
        const int v = t >> 3, l = (t >> 1) & 3, j = t & 1;
        float s = 0.f;
        for (int q = 0; q < 3; ++q)
            s += pro_inv[v * 12 + l * 3 + q] * cam_enc[q * 2 + j];
        pe[v][l][j] = s;
    }
    if (t >= 32 && t < 32 + NVIEW * NKPT) {
        const int u = t - 32;
        vs[u / NKPT][u % NKPT] = valid[((size_t)(u / NKPT) * NPRED + n) * NKPT + (u % NKPT)];
    }
    __syncthreads();

    // kp_nodes: enc[f=4i+l] = pooled[2i]*pe[l][0] + pooled[2i+1]*pe[l][1]
    for (int e = t; e < NKPT * 128; e += 128) {
        const int k = e >> 7, f = e & 127;
        const int i = f >> 2, l = f & 3;
        const float* p0 = pooled + ((size_t)(0 * NPRED + n) * NKPT + k) * OUTC;
        const float* p1 = pooled + ((size_t)(1 * NPRED + n) * NKPT + k) * OUTC;
        const float e0 = p0[2 * i] * pe[0][l][0] + p0[2 * i + 1] * pe[0][l][1];
        const float e1 = p1[2 * i] * pe[1][l][0] + p1[2 * i + 1] * pe[1][l][1];
        const float v0 = vs[0][k], v1 = vs[1][k];
        float vv = v0 + v1;
        vv = (vv == 0.f) ? 0.1f : vv;
        xb[k][f] = (e0 * v0 + e1 * v1) / vv;
    }
    __syncthreads();

    const float* gw[4]  = {g0, g1, g2, g3};
    const int    din[4]  = {128, 64, 32, 16};
    const int    dout[4] = {64, 32, 16, 3};

    for (int L = 0; L < 4; ++L) {
        const int fi = din[L], fo = dout[L];
        // t = LG @ x  (sparse: <=5 neighbors per node)
        for (int e = t; e < NKPT * fi; e += 128) {
            const int i = e / fi, f = e % fi;
            const unsigned m = kAdj[i];
            float s = 0.f;
            for (int j = 0; j < NKPT; ++j)
                if ((m >> j) & 1u) s += kDinv[j] * xb[j][f];
            tb[i][f] = kDinv[i] * s;
        }
        __syncthreads();
        // x = relu(t[k] @ g[k])  per keypoint
        for (int e = t; e < NKPT * fo; e += 128) {
            const int k = e / fo, h = e % fo;
            const float* gk = gw[L] + (size_t)k * fi * fo + h;
            float s = 0.f;
            for (int f = 0; f < fi; ++f)
                s += tb[k][f] * gk[(size_t)f * fo];
            xb[k][h] = fmaxf(s, 0.f);
        }
        __syncthreads();
    }

    for (int e = t; e < NKPT * 3; e += 128) {
        const int k = e / 3, h = e % 3;
        out[((size_t)n * NKPT + k) * 3 + h] = xb[k][h];
    }
}

extern "C" void kernel_launch(void* const* d_in, const int* in_sizes, int n_in,
                              void* d_out, int out_size, void* d_ws, size_t ws_size,
                              hipStream_t stream) {
    (void)in_sizes; (void)n_in; (void)out_size; (void)ws_size;
    const float* pre_feature = (const float*)d_in[0];
    const int*   pred_xy     = (const int*)  d_in[1];
    const float* valid       = (const float*)d_in[2];
    const float* pro_inv     = (const float*)d_in[3];
    const float* conv_w      = (const float*)d_in[4];
    const float* conv_b      = (const float*)d_in[5];
    const float* bn_gamma    = (const float*)d_in[6];
    const float* bn_beta     = (const float*)d_in[7];
    const float* bn_mean     = (const float*)d_in[8];
    const float* bn_var      = (const float*)d_in[9];
    const float* cam_enc     = (const float*)d_in[10];
    const float* g0          = (const float*)d_in[11];
    const float* g1          = (const float*)d_in[12];
    const float* g2          = (const float*)d_in[13];
    const float* g3          = (const float*)d_in[14];

    float* pooled = (float*)d_ws;            // V*N*KPT*64 = 19200 floats

    kp_conv_pool<<<dim3(NVIEW * NPRED * NKPT), dim3(128), 0, stream>>>(
        pre_feature, pred_xy, conv_w, conv_b, bn_gamma, bn_beta, bn_mean, bn_var,
        pooled);

    kp_gnn<<<dim3(NPRED), dim3(128), 0, stream>>>(
        pooled, valid, pro_inv, cam_enc, g0, g1, g2, g3, (float*)d_out);
}